// GCN_raw_mean_46729244181070
// MI455X (gfx1250) — compile-verified
//
#include <hip/hip_runtime.h>
#include <hip/hip_bf16.h>
#include <math.h>

// ---------------------------------------------------------------------------
// GCN (3x GraphConv + BN + ReLU, mean-pool, MLP head) for MI455X / gfx1250.
// Edge phase: float4 gathers + global_atomic_add_f32 scatter (memory bound;
// agg/h buffers fit in the 192MB L2). Node phase: exact-fp32 WMMA
// (v_wmma_f32_16x16x4_f32), one wave32 per 16 nodes, K split statically into
// the agg@w_rel half and the x@w_root half (F_IN % 4 == 0 guarantees every
// 4-wide K chunk lies wholly in one half -> no runtime selects, and each
// lane's A pair is one contiguous b64 load).
// ---------------------------------------------------------------------------

typedef __attribute__((ext_vector_type(2))) float v2f;
typedef __attribute__((ext_vector_type(8))) float v8f;

#define EPS 1e-5f

// ------------------------------ utility ------------------------------------
__global__ void zero_f32_kernel(float* __restrict__ p, int n) {
    int i = blockIdx.x * blockDim.x + threadIdx.x;
    if (i < n) p[i] = 0.0f;
}

// --------------------------- edge scatter-add -------------------------------
// agg[dst] += ew * x[src], F features per edge, 4 features per thread.
template <int F>
__global__ void edge_scatter_kernel(const float* __restrict__ x,
                                    const int*   __restrict__ src,
                                    const int*   __restrict__ dst,
                                    const float* __restrict__ ew,
                                    float* __restrict__ agg, int e) {
    constexpr int QUADS = F / 4;
    long long t = (long long)blockIdx.x * blockDim.x + threadIdx.x;
    long long edge = t / QUADS;
    int f4 = (int)(t % QUADS) * 4;
    if (edge >= e) return;
    int s = src[edge];
    int d = dst[edge];
    float w = ew[edge];
    const float4 v = *reinterpret_cast<const float4*>(x + (long long)s * F + f4);
    float* ap = agg + (long long)d * F + f4;
    atomicAdd(ap + 0, v.x * w);
    atomicAdd(ap + 1, v.y * w);
    atomicAdd(ap + 2, v.z * w);
    atomicAdd(ap + 3, v.w * w);
}

// ------------------------ node update via fp32 WMMA -------------------------
// out = relu(bn(agg @ w_rel + x @ w_root + bias))
//
// A frag (16x4 f32): lanes 0-15 hold K = k0+{0,1}, lanes 16-31 K = k0+{2,3},
//                    row M = lane & 15.  (two contiguous floats -> one b64)
// B frag (4x16 f32): mirrored, col N = lane & 15.
// C/D   (16x16 f32): element [M = r + 8*(lane>>4)][N = lane & 15] in VGPR r.
template <int F_IN, int F_OUT>
__global__ void node_update_wmma_kernel(const float* __restrict__ agg,   // [N,F_IN]
                                        const float* __restrict__ xin,   // [N,F_IN]
                                        const float* __restrict__ w_rel, // [F_IN,F_OUT]
                                        const float* __restrict__ w_root,// [F_IN,F_OUT]
                                        const float* __restrict__ bias,  // [F_OUT]
                                        const float* __restrict__ bn_g,
                                        const float* __restrict__ bn_b,
                                        const float* __restrict__ bn_m,
                                        const float* __restrict__ bn_v,
                                        float* __restrict__ out,         // [N,F_OUT]
                                        int n) {
    constexpr int NT = F_OUT / 16;
    static_assert(F_IN % 4 == 0, "K chunks must not straddle the agg|x seam");

    const int lane  = threadIdx.x & 31;
    const int wave  = blockIdx.x * (blockDim.x >> 5) + (threadIdx.x >> 5);
    const int node0 = wave * 16;
    if (node0 >= n) return;                 // wave-uniform: EXEC stays all-1s

    const int  row   = lane & 15;           // M for A, N for B
    const int  khalf = (lane >> 4) * 2;     // 0 or 2
    const int  node  = node0 + row;
    const bool vld   = (node < n);
    const float amsk = vld ? 1.0f : 0.0f;   // zero ragged rows (tail wave only)
    const long long nr = (long long)(vld ? node : (n - 1)) * F_IN;

    v8f acc[NT];
    #pragma unroll
    for (int t = 0; t < NT; ++t)
        acc[t] = (v8f){0.f, 0.f, 0.f, 0.f, 0.f, 0.f, 0.f, 0.f};

    // ---- first half: agg @ w_rel ----
    #pragma unroll
    for (int k0 = 0; k0 < F_IN; k0 += 4) {
        const int ka = k0 + khalf;          // ka even -> 8B aligned pair load
        v2f A = *reinterpret_cast<const v2f*>(agg + nr + ka);
        A[0] *= amsk; A[1] *= amsk;
        #pragma unroll
        for (int t = 0; t < NT; ++t) {
            const int col = t * 16 + row;
            v2f B;
            B[0] = w_rel[(ka + 0) * F_OUT + col];
            B[1] = w_rel[(ka + 1) * F_OUT + col];
            acc[t] = __builtin_amdgcn_wmma_f32_16x16x4_f32(
                false, A, false, B, (short)0, acc[t], false, false);
        }
    }
    // ---- second half: x @ w_root ----
    #pragma unroll
    for (int k0 = 0; k0 < F_IN; k0 += 4) {
        const int ka = k0 + khalf;
        v2f A = *reinterpret_cast<const v2f*>(xin + nr + ka);
        A[0] *= amsk; A[1] *= amsk;
        #pragma unroll
        for (int t = 0; t < NT; ++t) {
            const int col = t * 16 + row;
            v2f B;
            B[0] = w_root[(ka + 0) * F_OUT + col];
            B[1] = w_root[(ka + 1) * F_OUT + col];
            acc[t] = __builtin_amdgcn_wmma_f32_16x16x4_f32(
                false, A, false, B, (short)0, acc[t], false, false);
        }
    }

    // ---- epilogue: bias + batchnorm (eval) + relu, store per C/D layout ----
    const int mbase = 8 * (lane >> 4);
    if (node0 + 16 <= n) {                  // wave-uniform fast path (no preds)
        #pragma unroll
        for (int t = 0; t < NT; ++t) {
            const int col  = t * 16 + row;
            const float sc = bn_g[col] * rsqrtf(bn_v[col] + EPS);
            const float sh = bn_b[col] - bn_m[col] * sc;
            const float bs = bias[col];
            #pragma unroll
            for (int r = 0; r < 8; ++r) {
                const long long nd = node0 + mbase + r;
                float val = fmaf(acc[t][r] + bs, sc, sh);
                out[nd * F_OUT + col] = fmaxf(val, 0.f);
            }
        }
    } else {                                // single ragged tail wave
        #pragma unroll
        for (int t = 0; t < NT; ++t) {
            const int col  = t * 16 + row;
            const float sc = bn_g[col] * rsqrtf(bn_v[col] + EPS);
            const float sh = bn_b[col] - bn_m[col] * sc;
            const float bs = bias[col];
            #pragma unroll
            for (int r = 0; r < 8; ++r) {
                const int nd = node0 + mbase + r;
                if (nd < n) {
                    float val = fmaf(acc[t][r] + bs, sc, sh);
                    out[(long long)nd * F_OUT + col] = fmaxf(val, 0.f);
                }
            }
        }
    }
}

// ------------------------------- pooling ------------------------------------
__global__ void pool_sum_kernel(const float* __restrict__ h,     // [N,64]
                                const int*   __restrict__ batch, // [N]
                                float* __restrict__ sums,        // [G,64]
                                float* __restrict__ cnts,        // [G]
                                int n) {
    long long i = (long long)blockIdx.x * blockDim.x + threadIdx.x;
    int node = (int)(i >> 6);
    int f = (int)(i & 63);
    if (node >= n) return;
    int g = batch[node];
    atomicAdd(&sums[g * 64 + f], h[(long long)node * 64 + f]);
    if (f == 0) atomicAdd(&cnts[g], 1.0f);
}

// ------------------------------- MLP head -----------------------------------
// one block per graph: pooled -> fc1 -> bn -> relu -> fc4 -> log_softmax
__global__ void head_kernel(const float* __restrict__ sums, const float* __restrict__ cnts,
                            const float* __restrict__ fc1_w, const float* __restrict__ fc1_b,
                            const float* __restrict__ bnf_g, const float* __restrict__ bnf_b,
                            const float* __restrict__ bnf_m, const float* __restrict__ bnf_v,
                            const float* __restrict__ fc4_w, const float* __restrict__ fc4_b,
                            float* __restrict__ out) {
    __shared__ float pooled[64];
    __shared__ float z16[16];
    __shared__ float z2[2];
    const int g = blockIdx.x;
    const int t = threadIdx.x;

    const float inv = 1.0f / fmaxf(cnts[g], 1.0f);
    if (t < 64) pooled[t] = sums[g * 64 + t] * inv;
    __syncthreads();

    if (t < 16) {
        float a = fc1_b[t];
        #pragma unroll
        for (int k = 0; k < 64; ++k) a = fmaf(pooled[k], fc1_w[k * 16 + t], a);
        const float sc = bnf_g[t] * rsqrtf(bnf_v[t] + EPS);
        a = fmaf(a - bnf_m[t], sc, bnf_b[t]);
        z16[t] = fmaxf(a, 0.f);
    }
    __syncthreads();

    if (t < 2) {
        float a = fc4_b[t];
        #pragma unroll
        for (int k = 0; k < 16; ++k) a = fmaf(z16[k], fc4_w[k * 2 + t], a);
        z2[t] = a;
    }
    __syncthreads();

    if (t < 2) {
        const float m   = fmaxf(z2[0], z2[1]);
        const float lse = m + logf(expf(z2[0] - m) + expf(z2[1] - m));
        out[g * 2 + t] = z2[t] - lse;
    }
}

// ------------------------------- launcher -----------------------------------
static inline int cdiv(long long a, long long b) { return (int)((a + b - 1) / b); }

extern "C" void kernel_launch(void* const* d_in, const int* in_sizes, int n_in,
                              void* d_out, int out_size, void* d_ws, size_t ws_size,
                              hipStream_t stream) {
    const int N = in_sizes[0] / 4;   // x is [N,4]
    const int E = in_sizes[2];       // edge_weight is [E]
    const int G = out_size / 2;      // output is [G,2]

    const float* x       = (const float*)d_in[0];
    const int*   eidx    = (const int*)  d_in[1];   // [2,E] flat
    const int*   src     = eidx;
    const int*   dst     = eidx + E;
    const float* ew      = (const float*)d_in[2];
    const int*   batch   = (const int*)  d_in[3];
    const float* w1_rel  = (const float*)d_in[4];
    const float* w1_root = (const float*)d_in[5];
    const float* b1      = (const float*)d_in[6];
    const float* w2_rel  = (const float*)d_in[7];
    const float* w2_root = (const float*)d_in[8];
    const float* b2      = (const float*)d_in[9];
    const float* w3_rel  = (const float*)d_in[10];
    const float* w3_root = (const float*)d_in[11];
    const float* b3      = (const float*)d_in[12];
    const float* bn1_g = (const float*)d_in[13], *bn1_b = (const float*)d_in[14];
    const float* bn1_m = (const float*)d_in[15], *bn1_v = (const float*)d_in[16];
    const float* bn2_g = (const float*)d_in[17], *bn2_b = (const float*)d_in[18];
    const float* bn2_m = (const float*)d_in[19], *bn2_v = (const float*)d_in[20];
    const float* bn3_g = (const float*)d_in[21], *bn3_b = (const float*)d_in[22];
    const float* bn3_m = (const float*)d_in[23], *bn3_v = (const float*)d_in[24];
    const float* bnf_g = (const float*)d_in[25], *bnf_b = (const float*)d_in[26];
    const float* bnf_m = (const float*)d_in[27], *bnf_v = (const float*)d_in[28];
    const float* fc1_w = (const float*)d_in[29], *fc1_b = (const float*)d_in[30];
    const float* fc4_w = (const float*)d_in[31], *fc4_b = (const float*)d_in[32];
    float* out = (float*)d_out;

    // workspace layout (fp32)
    float* ws   = (float*)d_ws;
    float* agg  = ws;                        // N*32 max (reused per layer)
    float* h1   = agg  + (size_t)N * 32;     // N*16
    float* h2   = h1   + (size_t)N * 16;     // N*32
    float* h3   = h2   + (size_t)N * 32;     // N*64
    float* sums = h3   + (size_t)N * 64;     // G*64
    float* cnts = sums + (size_t)G * 64;     // G
    (void)ws_size; (void)n_in;

    const int ZB = 256;
    const int waves_needed = cdiv(N, 16);
    const int nu_blocks = cdiv(waves_needed, 8);   // 8 waves / 256-thread block

    // ---------------- layer 1: 4 -> 16 ----------------
    zero_f32_kernel<<<cdiv((long long)N * 4, ZB), ZB, 0, stream>>>(agg, N * 4);
    edge_scatter_kernel<4><<<cdiv((long long)E * 1, ZB), ZB, 0, stream>>>(x, src, dst, ew, agg, E);
    node_update_wmma_kernel<4, 16><<<nu_blocks, 256, 0, stream>>>(
        agg, x, w1_rel, w1_root, b1, bn1_g, bn1_b, bn1_m, bn1_v, h1, N);

    // ---------------- layer 2: 16 -> 32 ----------------
    zero_f32_kernel<<<cdiv((long long)N * 16, ZB), ZB, 0, stream>>>(agg, N * 16);
    edge_scatter_kernel<16><<<cdiv((long long)E * 4, ZB), ZB, 0, stream>>>(h1, src, dst, ew, agg, E);
    node_update_wmma_kernel<16, 32><<<nu_blocks, 256, 0, stream>>>(
        agg, h1, w2_rel, w2_root, b2, bn2_g, bn2_b, bn2_m, bn2_v, h2, N);

    // ---------------- layer 3: 32 -> 64 ----------------
    zero_f32_kernel<<<cdiv((long long)N * 32, ZB), ZB, 0, stream>>>(agg, N * 32);
    edge_scatter_kernel<32><<<cdiv((long long)E * 8, ZB), ZB, 0, stream>>>(h2, src, dst, ew, agg, E);
    node_update_wmma_kernel<32, 64><<<nu_blocks, 256, 0, stream>>>(
        agg, h2, w3_rel, w3_root, b3, bn3_g, bn3_b, bn3_m, bn3_v, h3, N);

    // ---------------- mean pool + head ----------------
    zero_f32_kernel<<<cdiv(G * 65, ZB), ZB, 0, stream>>>(sums, G * 65); // sums + cnts contiguous
    pool_sum_kernel<<<cdiv((long long)N * 64, ZB), ZB, 0, stream>>>(h3, batch, sums, cnts, N);
    head_kernel<<<G, 64, 0, stream>>>(sums, cnts, fc1_w, fc1_b,
                                      bnf_g, bnf_b, bnf_m, bnf_v,
                                      fc4_w, fc4_b, out);
}